// CINLayer_10522669875581
// MI455X (gfx1250) — compile-verified
//
#include <hip/hip_runtime.h>
#include <hip/hip_bf16.h>
#include <stdint.h>

typedef __attribute__((ext_vector_type(2))) float v2f;
typedef __attribute__((ext_vector_type(8))) float v8f;
typedef unsigned int v4u __attribute__((ext_vector_type(4)));
typedef int v4i __attribute__((ext_vector_type(4)));
typedef int v8i __attribute__((ext_vector_type(8)));

#define B_N   2048
#define F_N   40
#define D_N   32
#define O_N   64
#define K1_N  1600   // F*F
#define K2_N  2560   // O0*F
#define NDW   8      // d-channels per workgroup (one per wave)
#define KT    16     // K-tile per TDM stage
#define WROW  9      // 8 d-floats + 1 pad dword (TDM pad_enable) -> conflict-free B reads
#define MB    32     // batch rows per workgroup (two 16-row WMMA tiles per wave)
#define RPAD  (MB + 1)

// Low 32 bits of a generic pointer into the LDS aperture == LDS byte offset.
__device__ __forceinline__ uint32_t lds_off(const void* p) {
  return (uint32_t)(uintptr_t)p;
}

// ---------------------------------------------------------------------------
// TDM: DMA one W tile [KT hf] x [64 o] x [8 d] into LDS (padded 9-dword rows).
//   dim0 (x): 8 d-elements, contiguous (32B)            tile_dim0 = 8
//   dim1 (y): 64 o, stride K*32 elements                tile_dim1 = 64
//   dim2 (z): 16 hf, stride 32 elements                 tile_dim2 = 16
// pad: +1 dword after every 8 dwords -> LDS pitch 9 dwords per (hf,o) chunk.
// ---------------------------------------------------------------------------
__device__ __forceinline__ void tdm_issue_wtile(const float* __restrict__ Wg,
                                                int K, int kbase, int d0,
                                                uint32_t lds_byte_addr) {
  uint64_t ga = (uint64_t)(uintptr_t)Wg + (((uint64_t)kbase * D_N + (uint64_t)d0) << 2);
  uint64_t stride_o = (uint64_t)K * D_N;          // elements between o-lines

  v4u g0;
  g0.x = 1u;                                      // count=1, user-mode load
  g0.y = lds_byte_addr;                           // lds_addr
  g0.z = (uint32_t)(ga & 0xffffffffu);            // global_addr[31:0]
  g0.w = (uint32_t)((ga >> 32) & 0x01ffffffu)     // global_addr[56:32]
       | (2u << 30);                              // type = 2 ("image")

  v8i g1;
  g1[0] = (int)((2u << 16)      // data_size = 4B
              | (1u << 20)      // pad_enable
              | (2u << 22)      // pad_interval: every 8 dwords
              | (0u << 25));    // pad_amount: 1 dword
  g1[1] = (int)(8u << 16);                        // tensor_dim0 = 8 (low16)
  g1[2] = (int)(64u << 16);                       // dim0 hi=0 | tensor_dim1 = 64 (low16)
  g1[3] = (int)(8u << 16);                        // dim1 hi=0 | tile_dim0 = 8
  g1[4] = (int)(64u | (16u << 16));               // tile_dim1 = 64 | tile_dim2 = 16
  g1[5] = (int)(uint32_t)(stride_o & 0xffffffffu);          // tensor_dim0_stride lo
  g1[6] = (int)((uint32_t)((stride_o >> 32) & 0xffffu)      // stride0 hi16
              | (32u << 16));                               // tensor_dim1_stride = 32 (lo16)
  g1[7] = 0;                                                // stride1 hi

  v4i g2 = {16, 0, 0, 0};                         // tensor_dim2 = 16, no iterate
  v4i g3 = {0, 0, 0, 0};

#if __clang_major__ >= 23
  v8i g4 = {0, 0, 0, 0, 0, 0, 0, 0};
  __builtin_amdgcn_tensor_load_to_lds(g0, g1, g2, g3, g4, 0);
#else
  __builtin_amdgcn_tensor_load_to_lds(g0, g1, g2, g3, 0);
#endif
}

// ---------------------------------------------------------------------------
// One CIN layer. Per wave (= one d-channel): y[0:32, 0:64] = relu(z @ W^T + b)
// as a 2 (M) x 4 (N) grid of 16x16 WMMA tiles; the B fragment is shared by
// both M tiles (1.5 LDS dwords per WMMA).
// A 16x4 f32 frag: a.x = A[m][2*lh], a.y = A[m][2*lh+1]
// B  4x16 f32 frag: b.x = B[2*lh][m], b.y = B[2*lh+1][m]
// C/D 16x16 f32: vgpr i -> row (i + 8*lh), col m.
// ---------------------------------------------------------------------------
__device__ __forceinline__ void cin_layer(const float* __restrict__ Wg,
                                          const float* __restrict__ bias,
                                          int K,
                                          const float* xkA, int xk_stride, // [NDW] slices for z-staging
                                          const float* x0A,               // xs base, stride F_N*RPAD
                                          float (*zs)[KT][RPAD],          // [NDW][KT][RPAD]
                                          float (*wt)[KT][O_N][WROW],     // [2][KT][64][9]
                                          float* yout,                    // own slice, 64 x RPAD
                                          int tid, int dl, int lh, int m, int d0) {
  const int row_t = tid & 31;        // this thread's fixed row for z-staging
  const int hf8_t = tid >> 5;        // covers hf offsets hf8_t and hf8_t+8

  v8f acc[2][4];
#pragma unroll
  for (int r = 0; r < 2; ++r)
#pragma unroll
    for (int t = 0; t < 4; ++t)
      acc[r][t] = (v8f){0.f, 0.f, 0.f, 0.f, 0.f, 0.f, 0.f, 0.f};

  const int nstages = K / KT;
  if (dl == 0) tdm_issue_wtile(Wg, K, 0, d0, lds_off(&wt[0][0][0][0]));

  for (int s = 0; s < nstages; ++s) {
    const int cur = s & 1;
    if (dl == 0) {
      if (s + 1 < nstages) {
        // prefetch next tile into the other buffer (consumers of it passed
        // the bottom barrier of stage s-1 already)
        tdm_issue_wtile(Wg, K, (s + 1) * KT, d0, lds_off(&wt[cur ^ 1][0][0][0]));
        __builtin_amdgcn_s_wait_tensorcnt(1);   // tile s resident
      } else {
        __builtin_amdgcn_s_wait_tensorcnt(0);
      }
    }

    // Cooperative z-staging: z[hf][row] = xk[h][row]*x0[f][row], all 8 d.
    // (h,f) decode happens twice per stage per thread.
    {
      int hfA = s * KT + hf8_t;
      int hA = hfA / F_N, fA = hfA - F_N * hA;
      int hfB = hfA + 8;
      int hB = hfB / F_N, fB = hfB - F_N * hB;
#pragma unroll
      for (int q = 0; q < NDW; ++q) {
        const float* xkq = xkA + q * xk_stride;
        const float* x0q = x0A + q * (F_N * RPAD);
        zs[q][hf8_t    ][row_t] = xkq[hA * RPAD + row_t] * x0q[fA * RPAD + row_t];
        zs[q][hf8_t + 8][row_t] = xkq[hB * RPAD + row_t] * x0q[fB * RPAD + row_t];
      }
    }
    __syncthreads();   // tile + z visible to all waves

#pragma unroll
    for (int kk = 0; kk < KT / 4; ++kk) {
      const int kb = kk * 4 + 2 * lh;
      v2f a0, a1;
      a0.x = zs[dl][kb][m];           a0.y = zs[dl][kb + 1][m];
      a1.x = zs[dl][kb][16 + m];      a1.y = zs[dl][kb + 1][16 + m];
#pragma unroll
      for (int t = 0; t < 4; ++t) {
        v2f bf;
        bf.x = wt[cur][kb][t * 16 + m][dl];
        bf.y = wt[cur][kb + 1][t * 16 + m][dl];
        acc[0][t] = __builtin_amdgcn_wmma_f32_16x16x4_f32(
            false, a0, false, bf, (short)0, acc[0][t], false, false);
        acc[1][t] = __builtin_amdgcn_wmma_f32_16x16x4_f32(
            false, a1, false, bf, (short)0, acc[1][t], false, false);
      }
    }
    __syncthreads();   // everyone done with buf[cur] and zs before overwrite
  }

  // bias + relu, store transposed into LDS: yout[o][row]
#pragma unroll
  for (int r = 0; r < 2; ++r)
#pragma unroll
    for (int t = 0; t < 4; ++t) {
      int o = t * 16 + m;
      float bb = bias[o];
#pragma unroll
      for (int i = 0; i < 8; ++i) {
        int mm = r * 16 + (lh << 3) + i;
        float v = acc[r][t][i] + bb;
        yout[o * RPAD + mm] = v > 0.f ? v : 0.f;
      }
    }
}

// Sum this WG's 8 d-channels and write partials for the d-group.
__device__ __forceinline__ void reduce_part(const float (*ys)[O_N][RPAD],
                                            float* __restrict__ part,
                                            int dg, int b0row, int tid, int col0) {
  __syncthreads();   // all waves' yout written
#pragma unroll
  for (int r = 0; r < 8; ++r) {
    int idx = r * 256 + tid;   // 0..2047 -> (row, o)
    int mr  = idx >> 6;
    int o   = idx & 63;
    float sacc = 0.f;
#pragma unroll
    for (int q = 0; q < NDW; ++q) sacc += ys[q][o][mr];
    part[((size_t)dg * B_N + (size_t)(b0row + mr)) * 128 + col0 + o] = sacc;
  }
}

extern "C" __global__ void __launch_bounds__(256)
cin_fused_kernel(const float* __restrict__ x,
                 const float* __restrict__ W0, const float* __restrict__ bias0,
                 const float* __restrict__ W1, const float* __restrict__ bias1,
                 float* __restrict__ part) {
  __shared__ __align__(16)  float xs[NDW][F_N][RPAD];    // x slice:  [d][f][row]
  __shared__ __align__(16)  float x1s[NDW][O_N][RPAD];   // x1 slice: [d][o][row]
  __shared__ __align__(16)  float zsb[NDW][KT][RPAD];    // staged A (z) tile
  __shared__ __align__(128) float wt[2][KT][O_N][WROW];  // TDM double-buffered W tiles

  const int tid   = threadIdx.x;
  const int dl    = tid >> 5;          // wave id == local d-channel
  const int lane  = tid & 31;
  const int lh    = lane >> 4;
  const int m     = lane & 15;
  const int dg    = blockIdx.x;        // 0..3  d-group
  const int b0row = blockIdx.y * MB;   // 0..2016
  const int d0    = dg * NDW;

  // Stage x[b0:b0+32, 0:40, d0:d0+8] into xs[d][f][row] (float2 per lane).
#pragma unroll
  for (int i = 0; i < 20; ++i) {
    int idx = i * 256 + tid;           // 0..5119 float2 slots (32*40*8/2)
    int p   = idx >> 2;                // (row, f) pair, 0..1279
    int sub = idx & 3;
    int bl  = p / F_N;
    int f   = p % F_N;
    const float2* g = reinterpret_cast<const float2*>(
        x + ((size_t)(b0row + bl) * F_N + f) * D_N + d0) + sub;
    float2 v = *g;
    xs[sub * 2    ][f][bl] = v.x;
    xs[sub * 2 + 1][f][bl] = v.y;
  }
  __syncthreads();   // xs visible before first z-staging

  // Layer 1: z = outer(x, x), y1 = relu(z @ W0^T + b0)
  cin_layer(W0, bias0, K1_N,
            &xs[0][0][0], F_N * RPAD, &xs[0][0][0],
            zsb, wt, &x1s[dl][0][0], tid, dl, lh, m, d0);
  reduce_part(x1s, part, dg, b0row, tid, 0);

  // Layer 2: z = outer(x1, x), y2 = relu(z @ W1^T + b1); reuse x1s for y2
  cin_layer(W1, bias1, K2_N,
            &x1s[0][0][0], O_N * RPAD, &xs[0][0][0],
            zsb, wt, &x1s[dl][0][0], tid, dl, lh, m, d0);
  reduce_part(x1s, part, dg, b0row, tid, 64);
}

// Deterministic final reduction: fixed-order sums over d / d-groups.
extern "C" __global__ void __launch_bounds__(192)
cin_finalize_kernel(const float* __restrict__ x,
                    const float* __restrict__ part,
                    float* __restrict__ out) {
  const int b = blockIdx.x;
  const int j = threadIdx.x;
  if (j >= 168) return;
  float s = 0.f;
  if (j < 40) {
    const float* px = x + ((size_t)b * F_N + j) * D_N;
#pragma unroll
    for (int d = 0; d < D_N; ++d) s += px[d];
  } else {
    const int c = j - 40;  // 0..127
#pragma unroll
    for (int dg = 0; dg < 4; ++dg)
      s += part[((size_t)dg * B_N + b) * 128 + c];
  }
  out[(size_t)b * 168 + j] = s;
}

extern "C" void kernel_launch(void* const* d_in, const int* in_sizes, int n_in,
                              void* d_out, int out_size, void* d_ws, size_t ws_size,
                              hipStream_t stream) {
  (void)in_sizes; (void)n_in; (void)out_size; (void)ws_size;
  const float* x  = (const float*)d_in[0];
  const float* W0 = (const float*)d_in[1];
  const float* b0 = (const float*)d_in[2];
  const float* W1 = (const float*)d_in[3];
  const float* b1 = (const float*)d_in[4];
  float* part = (float*)d_ws;                 // [4][2048][128] fp32 = 4 MB
  float* out  = (float*)d_out;                // [2048][168] fp32

  dim3 grid(4, B_N / MB);                     // (d-group, b-tile)
  cin_fused_kernel<<<grid, 256, 0, stream>>>(x, W0, b0, W1, b1, part);
  cin_finalize_kernel<<<B_N, 192, 0, stream>>>(x, part, out);
}